// ACCELoss_80504866996279
// MI455X (gfx1250) — compile-verified
//
#include <hip/hip_runtime.h>
#include <math.h>

#define NROWS 65536
#define NCOLS 1000
#define NBINS 15
#define CPAD 1024                      // padded row length in floats
#define ROWBYTES (CPAD * 4)            // 4096 B per row buffer
#define BLOCK 256
#define WAVES_PER_WG (BLOCK / 32)      // 8 wave32s
#define NWG 512
#define HIST_FLOATS (NBINS * NCOLS)    // 15000
#define SMEM_BYTES (HIST_FLOATS * 4 + WAVES_PER_WG * 2 * ROWBYTES)  // 125536 B

#if defined(__HIP_DEVICE_COMPILE__) && defined(__gfx1250__)
#if __has_builtin(__builtin_amdgcn_global_load_async_to_lds_b128)
#define HAVE_ASYNC 1
#endif
#endif

typedef int v4i __attribute__((vector_size(16)));
typedef __attribute__((address_space(1))) v4i gv4i;   // global int4
typedef __attribute__((address_space(3))) v4i lv4i;   // LDS int4

// ---- CDNA5 async global->LDS copy (16B per lane per issue, ASYNCcnt) ----
__device__ __forceinline__ void async_copy_b128(const void* g, void* l) {
#ifdef HAVE_ASYNC
  __builtin_amdgcn_global_load_async_to_lds_b128(
      (gv4i*)g, (lv4i*)l, /*imm offset*/ 0, /*cpol*/ 0);
#else
  *(float4*)l = *(const float4*)g;     // synchronous fallback
#endif
}

__device__ __forceinline__ void wait_async_keep8() {
#ifdef HAVE_ASYNC
#if __has_builtin(__builtin_amdgcn_s_wait_asynccnt)
  __builtin_amdgcn_s_wait_asynccnt(8);
#else
  asm volatile("s_wait_asynccnt 8" ::: "memory");
#endif
#endif
}

__device__ __forceinline__ void wait_async_all() {
#ifdef HAVE_ASYNC
#if __has_builtin(__builtin_amdgcn_s_wait_asynccnt)
  __builtin_amdgcn_s_wait_asynccnt(0);
#else
  asm volatile("s_wait_asynccnt 0" ::: "memory");
#endif
#endif
}

// Issue one row (4000 B, padded to 4096 B) as 8 async B128 transfers per wave.
// Tail addresses are clamped to stay inside the logits allocation; the
// clamped bytes land in LDS slots for c >= 1000 which compute ignores.
__device__ __forceinline__ void issue_row(const float* __restrict__ logits, int row,
                                          int lane, float* lbuf, const char* gend) {
  const char* gbase = (const char*)(logits + (size_t)row * NCOLS) + (size_t)(lane * 16);
  char* lbase = (char*)lbuf + lane * 16;
#pragma unroll
  for (int k = 0; k < 8; ++k) {
    const char* ga = gbase + k * 512;
    if (ga + 16 > gend) ga = gend - 16;      // keep DMA in-bounds (16B aligned)
    async_copy_b128(ga, lbase + k * 512);
  }
}

__device__ __forceinline__ void process_row(const float* lbuf, int row, int lane,
                                            const int* __restrict__ labels,
                                            float* hist, float* __restrict__ gacc) {
  const float4* lb4 = (const float4*)lbuf;
  float4 v[8];
#pragma unroll
  for (int j = 0; j < 8; ++j) {
    const int k = lane + 32 * j;             // float4 chunk id; k<250 covers c<1000
    if (k < 250) {
      v[j] = lb4[k];
    } else {
      v[j].x = v[j].y = v[j].z = v[j].w = -INFINITY;
    }
  }

  // row max (wave32 shuffle reduce)
  float m = -INFINITY;
#pragma unroll
  for (int j = 0; j < 8; ++j)
    m = fmaxf(m, fmaxf(fmaxf(v[j].x, v[j].y), fmaxf(v[j].z, v[j].w)));
#pragma unroll
  for (int off = 16; off > 0; off >>= 1) m = fmaxf(m, __shfl_xor(m, off, 32));

  // exp and row sum (exp(-inf)=0 for padding lanes)
  float s = 0.0f;
#pragma unroll
  for (int j = 0; j < 8; ++j) {
    v[j].x = __expf(v[j].x - m);
    v[j].y = __expf(v[j].y - m);
    v[j].z = __expf(v[j].z - m);
    v[j].w = __expf(v[j].w - m);
    s += (v[j].x + v[j].y) + (v[j].z + v[j].w);
  }
#pragma unroll
  for (int off = 16; off > 0; off >>= 1) s += __shfl_xor(s, off, 32);

  const float inv = 1.0f / s;
  const int lab = labels[row];               // wave-uniform -> scalar load

#pragma unroll
  for (int j = 0; j < 8; ++j) {
    const int k = lane + 32 * j;
    if (k >= 250) continue;
    const int c0 = 4 * k;
    const float cf[4] = {v[j].x * inv, v[j].y * inv, v[j].z * inv, v[j].w * inv};
#pragma unroll
    for (int t = 0; t < 4; ++t) {
      const float c = cf[t];
      if (c > 0.0f) {                        // conf==0 -> dummy bin (discarded)
        int b = (int)ceilf(c * (float)NBINS) - 1;
        b = b < 0 ? 0 : (b > NBINS - 1 ? NBINS - 1 : b);
        atomicAdd(&hist[b * NCOLS + (c0 + t)], c);          // ds_add_f32
        if ((c0 + t) == lab) atomicAdd(&gacc[b * NCOLS + lab], 1.0f);
      }
    }
  }
}

__global__ void __launch_bounds__(BLOCK)
ece_hist_kernel(const float* __restrict__ logits, const int* __restrict__ labels,
                float* __restrict__ gconf, float* __restrict__ gacc) {
  extern __shared__ float smem[];
  float* hist = smem;                                        // 15000 floats
  const int tid = threadIdx.x;
  const int lane = tid & 31;
  const int wave = tid >> 5;
  float* bufA = smem + HIST_FLOATS + wave * 2 * CPAD;        // per-wave double buffer
  float* bufB = bufA + CPAD;

  for (int i = tid; i < HIST_FLOATS; i += BLOCK) hist[i] = 0.0f;
  __syncthreads();

  const int gw = blockIdx.x * WAVES_PER_WG + wave;           // global wave id
  const int stride = NWG * WAVES_PER_WG;                     // 4096 waves total
  const char* gend = (const char*)(logits + (size_t)NROWS * NCOLS);

  float* curbuf = bufA;
  float* nxtbuf = bufB;
  issue_row(logits, gw, lane, curbuf, gend);                 // prologue prefetch

  for (int row = gw; row < NROWS; row += stride) {
    const int nrow = row + stride;
    const bool pf = (nrow < NROWS);
    if (pf) {
      issue_row(logits, nrow, lane, nxtbuf, gend);           // prefetch next row
      wait_async_keep8();                                    // current row's 8 done
    } else {
      wait_async_all();
    }
    process_row(curbuf, row, lane, labels, hist, gacc);
    float* t = curbuf; curbuf = nxtbuf; nxtbuf = t;
  }

  __syncthreads();
  for (int i = tid; i < HIST_FLOATS; i += BLOCK) {
    const float v = hist[i];
    if (v != 0.0f) atomicAdd(&gconf[i], v);                  // merge to global
  }
}

__global__ void __launch_bounds__(BLOCK)
ece_zero_kernel(float* __restrict__ ws, int n) {
  const int i = blockIdx.x * BLOCK + threadIdx.x;
  if (i < n) ws[i] = 0.0f;
}

__global__ void __launch_bounds__(BLOCK)
ece_final_kernel(const float* __restrict__ gconf, const float* __restrict__ gacc,
                 float* __restrict__ out) {
  __shared__ float red[BLOCK];
  float s = 0.0f;
  for (int i = threadIdx.x; i < HIST_FLOATS; i += BLOCK)
    s += fabsf(gconf[i] - gacc[i]);
  red[threadIdx.x] = s;
  __syncthreads();
  for (int k = BLOCK / 2; k > 0; k >>= 1) {
    if (threadIdx.x < k) red[threadIdx.x] += red[threadIdx.x + k];
    __syncthreads();
  }
  if (threadIdx.x == 0)
    out[0] = red[0] * (1.0f / ((float)NROWS * (float)NCOLS));
}

extern "C" void kernel_launch(void* const* d_in, const int* in_sizes, int n_in,
                              void* d_out, int out_size, void* d_ws, size_t ws_size,
                              hipStream_t stream) {
  (void)in_sizes; (void)n_in; (void)out_size; (void)ws_size;
  const float* logits = (const float*)d_in[0];
  const int* labels   = (const int*)d_in[1];
  float* out   = (float*)d_out;
  float* gconf = (float*)d_ws;              // 15000 floats
  float* gacc  = gconf + HIST_FLOATS;       // 15000 floats

  const int zn = 2 * HIST_FLOATS;
  ece_zero_kernel<<<(zn + BLOCK - 1) / BLOCK, BLOCK, 0, stream>>>(gconf, zn);
  ece_hist_kernel<<<NWG, BLOCK, SMEM_BYTES, stream>>>(logits, labels, gconf, gacc);
  ece_final_kernel<<<1, BLOCK, 0, stream>>>(gconf, gacc, out);
}